// FraudGNNModel_33560874451283
// MI455X (gfx1250) — compile-verified
//
#include <hip/hip_runtime.h>
#include <math.h>

typedef float v2f __attribute__((ext_vector_type(2)));
typedef float v8f __attribute__((ext_vector_type(8)));

#define NEG_SLOPE 0.2f
#define GAT_EPS 1e-16f

// ---------------------------------------------------------------------------
// float atomic max via int/uint trick (works for mixed signs; init to -inf)
__device__ __forceinline__ void atomicMaxFloat(float* addr, float value) {
    if (value >= 0.0f)
        atomicMax((int*)addr, __float_as_int(value));
    else
        atomicMin((unsigned int*)addr, __float_as_uint(value));
}

// ---------------------------------------------------------------------------
// Scalar init
__global__ void k_zero_scalars(float* scal) {
    if (threadIdx.x == 0) { scal[0] = 0.f; scal[1] = 0.f; scal[2] = 0.f; scal[3] = 0.f; }
}

// Sum of edge_attr -> scal[0]
__global__ void k_reduce_sum(const float* __restrict__ v, int n, float* __restrict__ out) {
    __shared__ float sm[8];
    float s = 0.f;
    for (int i = blockIdx.x * blockDim.x + threadIdx.x; i < n; i += gridDim.x * blockDim.x)
        s += v[i];
    #pragma unroll
    for (int off = 16; off > 0; off >>= 1) s += __shfl_down(s, off, 32);
    if ((threadIdx.x & 31) == 0) sm[threadIdx.x >> 5] = s;
    __syncthreads();
    if (threadIdx.x == 0) {
        float t = 0.f;
        #pragma unroll
        for (int i = 0; i < 8; ++i) t += sm[i];
        atomicAdd(out, t);
    }
}

// scal[1] = ea_mean, scal[2] = c1 = dot(W_edge1, att_edge1), scal[3] = c2
__global__ void k_scalars(const float* __restrict__ We1, const float* __restrict__ ae1,
                          const float* __restrict__ We2, const float* __restrict__ ae2,
                          float* __restrict__ scal, float invE) {
    __shared__ float sm[64];
    int t = threadIdx.x;  // 64 threads
    sm[t] = We1[t] * ae1[t];
    __syncthreads();
    for (int s = 32; s > 0; s >>= 1) { if (t < s) sm[t] += sm[t + s]; __syncthreads(); }
    if (t == 0) {
        scal[2] = sm[0];
        scal[3] = We2[0] * ae2[0] + We2[1] * ae2[1];
        scal[1] = scal[0] * invE;
    }
}

// ---------------------------------------------------------------------------
// h1 = x @ W1  via V_WMMA_F32_16X16X4_F32. Block = 256 thr = 8 waves.
// Each block: 32 rows of x; each wave owns one 16x16 output tile
// (2 row-tiles x 4 col-tiles). Full-precision f32 WMMA, K stepped by 4.
#define XSTRIDE 132   // 128 + 4 pad: avoids 16-way LDS bank conflicts on A reads
#define WSTRIDE 72    // 64 + 8 pad: halves land on disjoint bank groups for B reads
__global__ __launch_bounds__(256) void k_gemm_xw1(const float* __restrict__ x,
                                                  const float* __restrict__ W,
                                                  float* __restrict__ h, int N) {
    __shared__ float ldsX[32 * XSTRIDE];
    __shared__ float ldsW[128 * WSTRIDE];
    const int tid  = threadIdx.x;
    const int row0 = blockIdx.x * 32;

    // Stage W1 (128x64) into LDS with padded row stride
    for (int i = tid; i < 128 * 16; i += 256) {          // 16 float4 per row
        int k = i >> 4, n4 = i & 15;
        ((float4*)(ldsW + k * WSTRIDE))[n4] = ((const float4*)(W + k * 64))[n4];
    }
    // Stage 32x128 tile of x
    for (int i = tid; i < 32 * 32; i += 256) {           // 32 float4 per row
        int r = i >> 5, c = i & 31;
        int gr = row0 + r;
        float4 v = make_float4(0.f, 0.f, 0.f, 0.f);
        if (gr < N) v = ((const float4*)(x + (size_t)gr * 128))[c];
        ((float4*)(ldsX + r * XSTRIDE))[c] = v;
    }
    __syncthreads();

    const int wave  = tid >> 5;
    const int lane  = tid & 31;
    const int rt    = wave >> 2;   // row tile 0..1
    const int ct    = wave & 3;    // col tile 0..3
    const int half  = lane >> 4;   // 0: K pair {0,1}; 1: K pair {2,3}
    const int idx16 = lane & 15;   // M for A-operand, N for B/C/D

    v8f acc = {};
    const float* xrow = ldsX + (rt * 16 + idx16) * XSTRIDE;
    const int    bcol = ct * 16 + idx16;
    #pragma unroll
    for (int k0 = 0; k0 < 128; k0 += 4) {
        const int ka = k0 + 2 * half;
        v2f a; a.x = xrow[ka];                     a.y = xrow[ka + 1];
        v2f b; b.x = ldsW[ka * WSTRIDE + bcol];    b.y = ldsW[(ka + 1) * WSTRIDE + bcol];
        acc = __builtin_amdgcn_wmma_f32_16x16x4_f32(false, a, false, b,
                                                    (short)0, acc, false, false);
    }
    // C/D layout: lane(0-15): M=vr,   N=lane ; lane(16-31): M=vr+8, N=lane-16
    const int col = ct * 16 + idx16;
    #pragma unroll
    for (int vr = 0; vr < 8; ++vr) {
        int grow = row0 + rt * 16 + (vr + 8 * half);
        if (grow < N) h[(size_t)grow * 64 + col] = acc[vr];
    }
}

// ---------------------------------------------------------------------------
// a_s[n] = h[n]·att_src, a_d[n] = h[n]·att_dst  (layer 1, F=64)
__global__ void k_node_logits1(const float* __restrict__ h, const float* __restrict__ att_s,
                               const float* __restrict__ att_d,
                               float* __restrict__ a_s, float* __restrict__ a_d, int N) {
    int n = blockIdx.x * blockDim.x + threadIdx.x;
    if (n >= N) return;
    const float4* hr  = (const float4*)(h + (size_t)n * 64);
    const float4* as4 = (const float4*)att_s;
    const float4* ad4 = (const float4*)att_d;
    float ss = 0.f, sd = 0.f;
    #pragma unroll
    for (int i = 0; i < 16; ++i) {
        float4 hv = hr[i], av = as4[i], dv = ad4[i];
        ss += hv.x * av.x + hv.y * av.y + hv.z * av.z + hv.w * av.w;
        sd += hv.x * dv.x + hv.y * dv.y + hv.z * dv.z + hv.w * dv.w;
    }
    a_s[n] = ss; a_d[n] = sd;
}

__global__ void k_init1(float* __restrict__ m, float* __restrict__ den,
                        float* __restrict__ agg, int N) {
    int n = blockIdx.x * blockDim.x + threadIdx.x;
    if (n >= N) return;
    m[n] = -INFINITY; den[n] = 0.f;
    float4* a = (float4*)(agg + (size_t)n * 64);
    #pragma unroll
    for (int i = 0; i < 16; ++i) a[i] = make_float4(0.f, 0.f, 0.f, 0.f);
}

__global__ void k_init2(float* __restrict__ m, float* __restrict__ den,
                        float* __restrict__ agg2, int N) {
    int n = blockIdx.x * blockDim.x + threadIdx.x;
    if (n >= N) return;
    m[n] = -INFINITY; den[n] = 0.f;
    agg2[2 * n] = 0.f; agg2[2 * n + 1] = 0.f;
}

// ---------------------------------------------------------------------------
// Edge pass 1: leaky_relu logit + segment max (atomic) — shared by both layers
__global__ void k_edge_logits(const int* __restrict__ src, const int* __restrict__ dst,
                              const float* __restrict__ eattr, const float* __restrict__ scal,
                              int cidx,
                              const float* __restrict__ a_s, const float* __restrict__ a_d,
                              float* __restrict__ ebuf, float* __restrict__ m, int E, int Et) {
    int e = blockIdx.x * blockDim.x + threadIdx.x;
    if (e >= Et) return;
    int s, d; float ea;
    if (e < E) { s = src[e]; d = dst[e]; ea = eattr[e]; }
    else       { s = e - E;  d = s;      ea = scal[1]; }   // self-loop, mean attr
    float z = a_s[s] + a_d[d] + ea * scal[cidx];
    float logit = (z > 0.f) ? z : NEG_SLOPE * z;
    ebuf[e] = logit;
    atomicMaxFloat(&m[d], logit);
}

// Edge pass 2: exp(logit - max) + segment sum
__global__ void k_edge_exp(const int* __restrict__ dst, float* __restrict__ ebuf,
                           const float* __restrict__ m, float* __restrict__ den, int E, int Et) {
    int e = blockIdx.x * blockDim.x + threadIdx.x;
    if (e >= Et) return;
    int d = (e < E) ? dst[e] : (e - E);
    float mm = m[d];
    if (!isfinite(mm)) mm = 0.f;
    float ex = __expf(ebuf[e] - mm);
    ebuf[e] = ex;
    atomicAdd(&den[d], ex);
}

// Edge pass 3 (layer 1): out[dst] += alpha * h[src], 64 feats, 32 lanes/edge
__global__ void k_edge_agg64(const int* __restrict__ src, const int* __restrict__ dst,
                             const float* __restrict__ ebuf, const float* __restrict__ den,
                             const float* __restrict__ h, float* __restrict__ agg,
                             int E, int Et) {
    long long t = (long long)blockIdx.x * blockDim.x + threadIdx.x;
    int e    = (int)(t >> 5);
    int lane = (int)(t & 31);
    if (e >= Et) return;
    int s, d;
    if (e < E) { s = src[e]; d = dst[e]; } else { s = e - E; d = s; }
    float alpha = ebuf[e] / (den[d] + GAT_EPS);
    float2 hv = *(const float2*)(h + (size_t)s * 64 + lane * 2);
    float* out = agg + (size_t)d * 64 + lane * 2;
    atomicAdd(out,     alpha * hv.x);
    atomicAdd(out + 1, alpha * hv.y);
}

// relu(agg + bias) in place (layer-1 output -> layer-2 input)
__global__ void k_relu_bias(float* __restrict__ agg, const float* __restrict__ bias, int N16) {
    int i = blockIdx.x * blockDim.x + threadIdx.x;
    if (i >= N16) return;
    float4 v = ((float4*)agg)[i];
    float4 b = ((const float4*)bias)[i & 15];
    v.x = fmaxf(v.x + b.x, 0.f); v.y = fmaxf(v.y + b.y, 0.f);
    v.z = fmaxf(v.z + b.z, 0.f); v.w = fmaxf(v.w + b.w, 0.f);
    ((float4*)agg)[i] = v;
}

// Layer-2 feature transform: hh2 = h2 @ W2 [64x2]; a_s2/a_d2 fused
__global__ void k_node2(const float* __restrict__ h2, const float* __restrict__ W2,
                        const float* __restrict__ as2, const float* __restrict__ ad2,
                        float* __restrict__ hh2, float* __restrict__ a_s,
                        float* __restrict__ a_d, int N) {
    int n = blockIdx.x * blockDim.x + threadIdx.x;
    if (n >= N) return;
    const float4* hr = (const float4*)(h2 + (size_t)n * 64);
    float o0 = 0.f, o1 = 0.f;
    #pragma unroll
    for (int i = 0; i < 16; ++i) {
        float4 hv = hr[i];
        const float* w = W2 + i * 8;   // rows 4i..4i+3, 2 cols each
        o0 += hv.x * w[0] + hv.y * w[2] + hv.z * w[4] + hv.w * w[6];
        o1 += hv.x * w[1] + hv.y * w[3] + hv.z * w[5] + hv.w * w[7];
    }
    hh2[2 * n] = o0; hh2[2 * n + 1] = o1;
    a_s[n] = o0 * as2[0] + o1 * as2[1];
    a_d[n] = o0 * ad2[0] + o1 * ad2[1];
}

// Edge pass 3 (layer 2): 2 features, 1 thread/edge
__global__ void k_edge_agg2(const int* __restrict__ src, const int* __restrict__ dst,
                            const float* __restrict__ ebuf, const float* __restrict__ den,
                            const float* __restrict__ hh2, float* __restrict__ agg2,
                            int E, int Et) {
    int e = blockIdx.x * blockDim.x + threadIdx.x;
    if (e >= Et) return;
    int s, d;
    if (e < E) { s = src[e]; d = dst[e]; } else { s = e - E; d = s; }
    float alpha = ebuf[e] / (den[d] + GAT_EPS);
    atomicAdd(&agg2[2 * d],     alpha * hh2[2 * s]);
    atomicAdd(&agg2[2 * d + 1], alpha * hh2[2 * s + 1]);
}

// + bias2, log_softmax over 2 classes
__global__ void k_final(const float* __restrict__ agg2, const float* __restrict__ bias2,
                        float* __restrict__ out, int N) {
    int n = blockIdx.x * blockDim.x + threadIdx.x;
    if (n >= N) return;
    float v0 = agg2[2 * n] + bias2[0];
    float v1 = agg2[2 * n + 1] + bias2[1];
    float mx = fmaxf(v0, v1);
    float ls = mx + __logf(__expf(v0 - mx) + __expf(v1 - mx));
    out[2 * n]     = v0 - ls;
    out[2 * n + 1] = v1 - ls;
}

// ---------------------------------------------------------------------------
extern "C" void kernel_launch(void* const* d_in, const int* in_sizes, int n_in,
                              void* d_out, int out_size, void* d_ws, size_t ws_size,
                              hipStream_t stream) {
    (void)n_in; (void)out_size; (void)ws_size;
    const float* x         = (const float*)d_in[0];
    const int*   ei        = (const int*)d_in[1];
    const float* eattr     = (const float*)d_in[2];
    const float* W1        = (const float*)d_in[3];
    const float* att_src1  = (const float*)d_in[4];
    const float* att_dst1  = (const float*)d_in[5];
    const float* W_edge1   = (const float*)d_in[6];
    const float* att_edge1 = (const float*)d_in[7];
    const float* bias1     = (const float*)d_in[8];
    const float* W2        = (const float*)d_in[9];
    const float* att_src2  = (const float*)d_in[10];
    const float* att_dst2  = (const float*)d_in[11];
    const float* W_edge2   = (const float*)d_in[12];
    const float* att_edge2 = (const float*)d_in[13];
    const float* bias2     = (const float*)d_in[14];

    const int N  = in_sizes[0] / 128;
    const int E  = in_sizes[1] / 2;
    const int Et = E + N;
    const int* src = ei;
    const int* dst = ei + E;

    // workspace layout (floats)
    float* ws   = (float*)d_ws;
    size_t off  = 0;
    float* h1   = ws + off; off += (size_t)N * 64;
    float* agg1 = ws + off; off += (size_t)N * 64;   // relu'd in place -> h2
    float* ebuf = ws + off; off += (size_t)Et;
    float* a_s  = ws + off; off += (size_t)N;
    float* a_d  = ws + off; off += (size_t)N;
    float* mbuf = ws + off; off += (size_t)N;
    float* den  = ws + off; off += (size_t)N;
    float* hh2  = ws + off; off += (size_t)N * 2;
    float* agg2 = ws + off; off += (size_t)N * 2;
    float* scal = ws + off; off += 4;                 // [sum, mean, c1, c2]

    const int B = 256;
    const unsigned gN   = (unsigned)((N + B - 1) / B);
    const unsigned gEt  = (unsigned)((Et + B - 1) / B);
    const unsigned gAgg = (unsigned)(((long long)Et * 32 + B - 1) / B);
    const unsigned gN16 = (unsigned)(((long long)N * 16 + B - 1) / B);

    // scalars: ea_mean, c1, c2
    k_zero_scalars<<<1, 32, 0, stream>>>(scal);
    k_reduce_sum<<<512, B, 0, stream>>>(eattr, E, &scal[0]);
    k_scalars<<<1, 64, 0, stream>>>(W_edge1, att_edge1, W_edge2, att_edge2, scal, 1.0f / (float)E);

    // layer 1
    k_gemm_xw1<<<(unsigned)((N + 31) / 32), B, 0, stream>>>(x, W1, h1, N);
    k_node_logits1<<<gN, B, 0, stream>>>(h1, att_src1, att_dst1, a_s, a_d, N);
    k_init1<<<gN, B, 0, stream>>>(mbuf, den, agg1, N);
    k_edge_logits<<<gEt, B, 0, stream>>>(src, dst, eattr, scal, 2, a_s, a_d, ebuf, mbuf, E, Et);
    k_edge_exp<<<gEt, B, 0, stream>>>(dst, ebuf, mbuf, den, E, Et);
    k_edge_agg64<<<gAgg, B, 0, stream>>>(src, dst, ebuf, den, h1, agg1, E, Et);
    k_relu_bias<<<gN16, B, 0, stream>>>(agg1, bias1, N * 16);

    // layer 2
    k_node2<<<gN, B, 0, stream>>>(agg1, W2, att_src2, att_dst2, hh2, a_s, a_d, N);
    k_init2<<<gN, B, 0, stream>>>(mbuf, den, agg2, N);
    k_edge_logits<<<gEt, B, 0, stream>>>(src, dst, eattr, scal, 3, a_s, a_d, ebuf, mbuf, E, Et);
    k_edge_exp<<<gEt, B, 0, stream>>>(dst, ebuf, mbuf, den, E, Et);
    k_edge_agg2<<<gEt, B, 0, stream>>>(src, dst, ebuf, den, hh2, agg2, E, Et);
    k_final<<<gN, B, 0, stream>>>(agg2, bias2, (float*)d_out, N);
}